// KNNGeometric_14972255994213
// MI455X (gfx1250) — compile-verified
//
#include <hip/hip_runtime.h>
#include <stddef.h>

// Problem constants (from reference): BS=8, H=W=48, C=8, K=20
constexpr int BSz = 8;
constexpr int HWz = 48 * 48;   // 2304
constexpr int Cz  = 8;
constexpr int Kz  = 20;
constexpr int NPIX = BSz * HWz; // 18432

typedef __attribute__((ext_vector_type(2))) float v2f;
typedef __attribute__((ext_vector_type(8))) float v8f;

// ---------------------------------------------------------------------------
// Kernel 1: per-pixel 20th-largest correlation value (exact, wave-cooperative)
// One wave per pixel; 32 waves (32 adjacent pixels) per block share an LDS
// transpose tile so all global reads are coalesced 128B lines.
// ---------------------------------------------------------------------------
__global__ __launch_bounds__(1024)
void knn_thresh_kernel(const float* __restrict__ corr, float* __restrict__ thresh)
{
    __shared__ float tile[32][33];            // +1 pad: conflict-free transpose
    const int lane = threadIdx.x & 31;
    const int wid  = threadIdx.x >> 5;        // wave id in block == pixel offset
    const int p0   = blockIdx.x * 32;         // first global pixel of block
    const int p    = p0 + wid;                // this wave's pixel (0..18431)
    const int b    = p0 / HWz;                // uniform per block (2304 % 32 == 0)
    const int pix  = p  - b * HWz;            // pixel within image
    const int pix0 = p0 - b * HWz;            // block's first image pixel
    const float* __restrict__ cb = corr + (size_t)b * HWz * HWz;

    const float NEG = -3.402823466e38f;
    // per-lane branchless sorted top-4 (descending) of its 72-element subset
    float t0 = NEG, t1 = NEG, t2 = NEG, t3 = NEG;

    for (int t = 0; t < HWz / 32; ++t) {      // 72 tiles of 32 hw-rows
        // coalesced: wave 'wid' loads hw-row (t*32+wid), 32 adjacent pixels
        float v = cb[(size_t)(t * 32 + wid) * HWz + (pix0 + lane)];
        __syncthreads();                      // protect previous tile reads
        tile[wid][lane] = v;
        __syncthreads();
        float x = tile[lane][wid];            // hw = t*32+lane of my pixel
        float m0 = fminf(t0, x); t0 = fmaxf(t0, x);   // bubble insert, 7 ops
        float m1 = fminf(t1, m0); t1 = fmaxf(t1, m0);
        float m2 = fminf(t2, m1); t2 = fmaxf(t2, m1);
        t3 = fmaxf(t3, m2);
    }

    // Extract top-20 of the union of 32 sorted-4 lists via 20 wave-max rounds.
    float thr = NEG;
    float lastpop = 0.0f;
    bool exhausted = false;
    for (int r = 0; r < Kz; ++r) {
        float m = t0;
        m = fmaxf(m, __shfl_xor(m, 16));
        m = fmaxf(m, __shfl_xor(m, 8));
        m = fmaxf(m, __shfl_xor(m, 4));
        m = fmaxf(m, __shfl_xor(m, 2));
        m = fmaxf(m, __shfl_xor(m, 1));
        thr = m;
        if (!exhausted && t0 == m) {          // owner lane pops its head
            lastpop = t0; t0 = t1; t1 = t2; t2 = t3; t3 = NEG;
            if (t0 == NEG) {
                // Rare exact refill (~1.5% of pixels): this lane contributed
                // >4 of the top-20; rescan its 72 values strictly below the
                // last popped value (all survivors are < lastpop). L2-hot.
                for (int t = 0; t < HWz / 32; ++t) {
                    float x = cb[(size_t)(t * 32 + lane) * HWz + pix];
                    x = (x < lastpop) ? x : NEG;
                    float m0 = fminf(t0, x); t0 = fmaxf(t0, x);
                    float m1 = fminf(t1, m0); t1 = fmaxf(t1, m0);
                    float m2 = fminf(t2, m1); t2 = fmaxf(t2, m1);
                    t3 = fmaxf(t3, m2);
                }
                exhausted = (t0 == NEG);
            }
        }
    }
    if (lane == 0) thresh[p] = thr;           // 20th largest == mask threshold
}

// ---------------------------------------------------------------------------
// Kernel 2: out[b] = lbl[b] (8x2304, zero-padded to 16 rows) x masked corr[b]
// using V_WMMA_F32_16X16X4_F32 (exact f32). corr is re-read from L2 (170MB
// fits the 192MB L2). One wave per 16-pixel tile; 8 waves/block share the
// batch's label matrix staged in LDS. The A matrix is staged with its padding
// rows 8..15 physically zeroed, so the inner loop has NO divergent loads and
// EXEC stays all-ones (WMMA requirement) with no saveexec/restore overhead.
// ---------------------------------------------------------------------------
__global__ __launch_bounds__(256)
void knn_wmma_kernel(const float* __restrict__ corr,
                     const float* __restrict__ lbl,
                     const float* __restrict__ thresh,
                     float* __restrict__ out)
{
    constexpr int RS = HWz + 2;               // 2306: even (8B align) & bank-spread
    __shared__ float slbl[16 * RS];           // 16 rows: 8 real + 8 zero (148KB)

    const int b        = blockIdx.x / 18;     // 18 blocks per batch, 144 tiles
    const int tilebase = (blockIdx.x % 18) * 8;
    const int wid  = threadIdx.x >> 5;
    const int lane = threadIdx.x & 31;
    const int lm   = lane & 15;
    const int half = lane >> 4;

    // stage this batch's labels (8 x 2304 f32) with padded row stride
    const float* __restrict__ lb = lbl + (size_t)b * Cz * HWz;
    for (int i = threadIdx.x; i < Cz * HWz; i += 256) {
        int r = i / HWz, c = i - r * HWz;
        slbl[r * RS + c] = lb[i];
    }
    // zero the padding rows 8..15 once (removes per-iteration exec masking)
    for (int i = Cz * RS + threadIdx.x; i < 16 * RS; i += 256) {
        slbl[i] = 0.0f;
    }
    __syncthreads();

    const int tile = tilebase + wid;          // 0..143
    const int pb   = tile * 16;               // pixel base of this wave's tile
    const float* __restrict__ cb = corr + (size_t)b * HWz * HWz + pb + lm;
    const float tv = thresh[b * HWz + pb + lm];

    const int arow = lm * RS;                 // all 16 rows valid (padded zeros)

    v8f acc0 = {0,0,0,0,0,0,0,0};
    v8f acc1 = {0,0,0,0,0,0,0,0};

    for (int k0 = 0; k0 < HWz; k0 += 8) {     // two 16x16x4 WMMAs per iter
        const int kA = k0 + 2 * half;         // ISA A/B f32 layout: lanes 0-15
        // unconditional A loads (ds_load_2addr_b64-friendly pair)
        v2f a0 = *(const v2f*)&slbl[arow + kA];
        v2f a1 = *(const v2f*)&slbl[arow + kA + 4];
        // B: masked corr, coalesced 64B per 16-lane row
        float b00 = cb[(size_t)(kA    ) * HWz];
        float b01 = cb[(size_t)(kA + 1) * HWz];
        float b10 = cb[(size_t)(kA + 4) * HWz];
        float b11 = cb[(size_t)(kA + 5) * HWz];
        v2f bv0 = { b00 >= tv ? b00 : 0.0f, b01 >= tv ? b01 : 0.0f };
        v2f bv1 = { b10 >= tv ? b10 : 0.0f, b11 >= tv ? b11 : 0.0f };
        // D = A*B + C, full f32
        acc0 = __builtin_amdgcn_wmma_f32_16x16x4_f32(
                   false, a0, false, bv0, (short)0, acc0, false, false);
        acc1 = __builtin_amdgcn_wmma_f32_16x16x4_f32(
                   false, a1, false, bv1, (short)0, acc1, false, false);
    }

    // C/D layout: lanes 0-15 VGPR r -> (M=r, N=lane); lanes 16-31 hold M=8..15
    // which are padding rows here, so only half 0 stores.
    if (half == 0) {
        float* ob = out + (size_t)b * Cz * HWz + pb + lm;
        #pragma unroll
        for (int r = 0; r < Cz; ++r)
            ob[(size_t)r * HWz] = acc0[r] + acc1[r];
    }
}

// ---------------------------------------------------------------------------
extern "C" void kernel_launch(void* const* d_in, const int* in_sizes, int n_in,
                              void* d_out, int out_size, void* d_ws, size_t ws_size,
                              hipStream_t stream) {
    const float* corr = (const float*)d_in[0];   // (8, 2304, 48, 48) f32
    const float* lbl  = (const float*)d_in[1];   // (8, 8, 48, 48)    f32
    float* thresh = (float*)d_ws;                // 18432 floats (72KB)
    float* outp   = (float*)d_out;               // (8, 8, 48, 48)    f32

    knn_thresh_kernel<<<NPIX / 32, 1024, 0, stream>>>(corr, thresh);
    knn_wmma_kernel<<<BSz * 18, 256, 0, stream>>>(corr, lbl, thresh, outp);
}